// TaylorMap_33603824124566
// MI455X (gfx1250) — compile-verified
//
#include <hip/hip_runtime.h>

typedef __attribute__((ext_vector_type(2))) float v2f;
typedef __attribute__((ext_vector_type(4))) float v4f;
typedef __attribute__((ext_vector_type(8))) float v8f;

#define NFEAT  16
#define NPOLY  969            // comb(19,3)
#define NCHUNK 243            // ceil(969/4) K-chunks of 4 for wmma 16x16x4
#define LDSW   (NCHUNK * 64)  // 15552 floats = 62208 bytes

// ---------------------------------------------------------------------------
// Generation-order feature stream (a permutation of the reference's rows):
//   q = 0        : 1.0
//   q = 1..16    : x[q-1]
//   q = 17..968  : for i = 0..15, for j = 0..i:
//                      x_i*x_j, then x_i*x_j*x_k for k = 0..j
// Every feature is <= 2 muls from xv[]; no degree-2 array must stay live.
// ---------------------------------------------------------------------------
struct FDesc { int ty; int i; int j; int k; };  // ty: 0=one 1=x 2=xx 3=xxx 4=pad

constexpr FDesc fdesc(int q) {
    if (q == 0)   return {0, 0, 0, 0};
    if (q <= 16)  return {1, q - 1, 0, 0};
    if (q >= NPOLY) return {4, 0, 0, 0};
    int p = q - 17;
    for (int i = 0; i < 16; ++i)
        for (int j = 0; j <= i; ++j) {
            if (p == 0) return {2, i, j, 0};
            p -= 1;
            if (p <= j) return {3, i, j, p};
            p -= (j + 1);
        }
    return {4, 0, 0, 0};
}

// Runtime map: stream position q -> reference row index (for staging W').
//   ref(b2(i,j))   = 17  + i(i+1)/2 + j
//   ref(b3(i,j,k)) = 153 + C(i+2,3) + j(j+1)/2 + k
__device__ __forceinline__ int refrow_rt(int q) {
    if (q == 0)   return 0;
    if (q <= 16)  return q;
    if (q >= NPOLY) return -1;
    int p = q - 17;
    int i = 0;
    while (p >= (i + 1) * (i + 4) / 2) { p -= (i + 1) * (i + 4) / 2; ++i; }
    int j = 0;
    while (p >= j + 2) { p -= (j + 2); ++j; }
    if (p == 0) return 17 + i * (i + 1) / 2 + j;
    return 153 + i * (i + 1) * (i + 2) / 6 + j * (j + 1) / 2 + (p - 1);
}

// Compile-time feature value (indices are literal constants after folding).
template <int Q>
__device__ __forceinline__ float fval(const float (&xv)[16]) {
    constexpr FDesc d = fdesc(Q);
    if (d.ty == 0) return 1.0f;
    else if (d.ty == 1) return xv[d.i];
    else if (d.ty == 2) return xv[d.i] * xv[d.j];
    else if (d.ty == 3) return xv[d.i] * xv[d.j] * xv[d.k];
    else return 0.0f;
}

// Straight-line chunk chain via template recursion: nothing to re-roll.
template <int C>
struct Chunks {
    static __device__ __forceinline__ void run(v8f& acc, const float (&xv)[16],
                                               const float* ldsLane, bool hi) {
        float f0 = fval<4 * C + 0>(xv);
        float f1 = fval<4 * C + 1>(xv);
        float f2 = fval<4 * C + 2>(xv);
        float f3 = fval<4 * C + 3>(xv);
        v2f A;
        A.x = hi ? f2 : f0;
        A.y = hi ? f3 : f1;
        v2f B = *(const v2f*)(ldsLane + C * 64);   // ds_load_b64, imm offset C*256
        acc = __builtin_amdgcn_wmma_f32_16x16x4_f32(
                false, A, false, B, (short)0, acc, false, false);
        Chunks<C + 1>::run(acc, xv, ldsLane, hi);
    }
};
template <>
struct Chunks<NCHUNK> {
    static __device__ __forceinline__ void run(v8f&, const float (&)[16],
                                               const float*, bool) {}
};

__global__ __launch_bounds__(256) void taylor_wmma_f32(
        const float* __restrict__ x, const float* __restrict__ W,
        float* __restrict__ out, int ntiles) {
    __shared__ float ldsW[LDSW];

    const int tid = threadIdx.x;

    // Stage W' (= permuted W + identity-residual fold) in WMMA B layout:
    // LDS float c*64 + l*2 + j holds column l & 15 of W' stream row 4c + 2*(l>>4) + j
    for (int e = tid; e < LDSW; e += 256) {
        int c   = e >> 6;
        int r   = e & 63;
        int l   = r >> 1;
        int j   = r & 1;
        int q   = 4 * c + ((l >> 4) << 1) + j;   // stream row
        int col = l & 15;
        int ref = refrow_rt(q);                  // reference row (or -1 pad)
        float v = (ref >= 0) ? W[ref * 16 + col] : 0.0f;
        if (ref == col + 1) v += 1.0f;           // fold residual: poly[:,1+n]==x[:,n]
        ldsW[e] = v;
    }
    __syncthreads();

    const int  lane = tid & 31;
    const int  wave = tid >> 5;
    const bool hi   = (lane >= 16);              // lane half picks K=2,3 vs K=0,1
    const int  tile = blockIdx.x * 8 + wave;
    if (tile >= ntiles) return;                  // wave-uniform (EXEC stays all-ones)
    const int base = tile * 16;
    const int s    = base + (lane & 15);         // both halves mirror samples 0..15

    float xv[16];
    #pragma unroll
    for (int qd = 0; qd < 4; ++qd) {
        v4f t = ((const v4f*)(x + s * 16))[qd];  // global_load_b128
        xv[4 * qd + 0] = t.x; xv[4 * qd + 1] = t.y;
        xv[4 * qd + 2] = t.z; xv[4 * qd + 3] = t.w;
    }

    v8f acc = {};
    Chunks<0>::run(acc, xv, &ldsW[lane << 1], hi);   // 243 chained f32 WMMAs

    // D layout: lane (0-15|16-31), VGPR r -> out[base + r + 8*hi][lane&15]
    float* op = out + (base + (hi ? 8 : 0)) * 16 + (lane & 15);
    #pragma unroll
    for (int r = 0; r < 8; ++r)
        op[r * 16] = acc[r];
}

extern "C" void kernel_launch(void* const* d_in, const int* in_sizes, int n_in,
                              void* d_out, int out_size, void* d_ws, size_t ws_size,
                              hipStream_t stream) {
    const float* x = (const float*)d_in[0];   // [BATCH, 16] f32
    const float* W = (const float*)d_in[1];   // [969, 16] f32
    float* out = (float*)d_out;               // [BATCH, 16] f32

    const int batch  = in_sizes[0] / NFEAT;   // 262144
    const int ntiles = batch / 16;            // 16384
    const int blocks = (ntiles + 7) / 8;      // 8 waves (tiles) per 256-thread block

    taylor_wmma_f32<<<blocks, 256, 0, stream>>>(x, W, out, ntiles);
}